// Model_64020782514183
// MI455X (gfx1250) — compile-verified
//
#include <hip/hip_runtime.h>
#include <math.h>

typedef __attribute__((ext_vector_type(2))) float v2f;
typedef __attribute__((ext_vector_type(8))) float v8f;

constexpr int N_NODES = 100000;
constexpr int N_EDGES = 1600000;
constexpr int IN_C = 128, HID_C = 64, OUT_C = 40;
constexpr float BN_EPS = 1e-5f;

// ---------------------------------------------------------------------------
// Small utility kernels
// ---------------------------------------------------------------------------
__global__ void k_zero(float* p, int n) {
    int i = blockIdx.x * blockDim.x + threadIdx.x;
    if (i < n) p[i] = 0.0f;
}

__global__ void k_degree(const int* __restrict__ dst, float* __restrict__ deg) {
    int e = blockIdx.x * blockDim.x + threadIdx.x;
    if (e < N_EDGES) atomicAdd(&deg[dst[e]], 1.0f);
}

__global__ void k_node_norm(const float* __restrict__ deg,
                            float* __restrict__ dis, float* __restrict__ sn) {
    int i = blockIdx.x * blockDim.x + threadIdx.x;
    if (i < N_NODES) {
        float d = deg[i] + 1.0f;
        dis[i] = rsqrtf(d);
        sn[i]  = 1.0f / d;
    }
}

__global__ void k_edge_norm(const int* __restrict__ src, const int* __restrict__ dst,
                            const float* __restrict__ dis, float* __restrict__ en) {
    int e = blockIdx.x * blockDim.x + threadIdx.x;
    if (e < N_EDGES) en[e] = dis[src[e]] * dis[dst[e]];
}

// agg[i,c] = h[i,c] * self_norm[i]   (also serves as the self-loop term init)
template <int C>
__global__ void k_init_scaled(const float* __restrict__ h, const float* __restrict__ sn,
                              float* __restrict__ out) {
    const int total = N_NODES * C;
    for (int i = blockIdx.x * blockDim.x + threadIdx.x; i < total;
         i += gridDim.x * blockDim.x) {
        out[i] = h[i] * sn[i / C];
    }
}

// agg[dst[e],c] += h[src[e],c] * edge_norm[e]
template <int C>
__global__ void k_scatter_add(const float* __restrict__ h, const int* __restrict__ src,
                              const int* __restrict__ dst, const float* __restrict__ en,
                              float* __restrict__ agg) {
    const int total = N_EDGES * C;   // <= 102.4M, fits int32
    for (int i = blockIdx.x * blockDim.x + threadIdx.x; i < total;
         i += gridDim.x * blockDim.x) {
        int e = i / C;
        int c = i - e * C;
        float v = h[(size_t)src[e] * C + c] * en[e];
        atomicAdd(&agg[(size_t)dst[e] * C + c], v);
    }
}

// ---------------------------------------------------------------------------
// BatchNorm (over nodes) + ReLU, C = 64
// ---------------------------------------------------------------------------
__global__ void k_bn_stats(const float* __restrict__ h, float* __restrict__ stats) {
    // blockDim = 256; stride multiple of 64 -> each thread's channel is fixed
    const int total = N_NODES * HID_C;
    float s = 0.0f, q = 0.0f;
    for (int i = blockIdx.x * 256 + threadIdx.x; i < total; i += gridDim.x * 256) {
        float v = h[i];
        s += v;
        q += v * v;
    }
    __shared__ float ls[HID_C], lq[HID_C];
    if (threadIdx.x < HID_C) { ls[threadIdx.x] = 0.0f; lq[threadIdx.x] = 0.0f; }
    __syncthreads();
    int c = threadIdx.x & (HID_C - 1);
    atomicAdd(&ls[c], s);
    atomicAdd(&lq[c], q);
    __syncthreads();
    if (threadIdx.x < HID_C) {
        atomicAdd(&stats[threadIdx.x], ls[threadIdx.x]);
        atomicAdd(&stats[HID_C + threadIdx.x], lq[threadIdx.x]);
    }
}

__global__ void k_bn_finalize(const float* __restrict__ stats,
                              const float* __restrict__ gamma,
                              const float* __restrict__ beta,
                              float* __restrict__ ss) {
    int c = threadIdx.x;
    if (c < HID_C) {
        float mu  = stats[c] * (1.0f / N_NODES);
        float var = stats[HID_C + c] * (1.0f / N_NODES) - mu * mu;
        float sc  = gamma[c] * rsqrtf(var + BN_EPS);
        ss[c]          = sc;
        ss[HID_C + c]  = beta[c] - mu * sc;
    }
}

__global__ void k_bn_apply_relu(float* __restrict__ h, const float* __restrict__ ss) {
    const int total = N_NODES * HID_C;
    for (int i = blockIdx.x * blockDim.x + threadIdx.x; i < total;
         i += gridDim.x * blockDim.x) {
        int c = i & (HID_C - 1);
        float v = fmaf(h[i], ss[c], ss[HID_C + c]);
        h[i] = fmaxf(v, 0.0f);
    }
}

// ---------------------------------------------------------------------------
// f32 WMMA GEMM: H[M,NOUT] = A[M,K] @ W[K,NOUT] + bias
// One wave owns a 16-row strip and all output columns (NT 16-wide tiles).
// Uses V_WMMA_F32_16X16X4_F32 (full f32 precision, matches reference).
// A(16x4) layout: lanes 0-15 -> K = k0,k0+1 ; lanes 16-31 -> K = k0+2,k0+3
// B(4x16) layout: same K split across lane halves; N = lane&15
// C/D: VGPR r -> row r (lanes 0-15) / row r+8 (lanes 16-31), N = lane&15
// ---------------------------------------------------------------------------
template <int K, int NOUT>
__global__ void k_gemm_wmma(const float* __restrict__ A, const float* __restrict__ W,
                            const float* __restrict__ bias, float* __restrict__ H,
                            int nRowBlocks) {
    constexpr int NT = (NOUT + 15) / 16;
    const int wave = blockIdx.x * (blockDim.x >> 5) + (threadIdx.x >> 5);
    if (wave >= nRowBlocks) return;      // whole-wave uniform exit; EXEC stays all-1s
    const int lane = threadIdx.x & 31;
    const int half = lane >> 4;          // 0: lanes 0-15, 1: lanes 16-31
    const int l15  = lane & 15;
    const int row  = wave * 16 + l15;    // A row (same for both halves)

    v8f acc[NT] = {};

    const float* arow = A + (size_t)row * K;
    for (int k0 = 0; k0 < K; k0 += 4) {
        const int kb = k0 + half * 2;
        v2f a;
        a.x = arow[kb];
        a.y = arow[kb + 1];
#pragma unroll
        for (int t = 0; t < NT; ++t) {
            const int col = t * 16 + l15;
            v2f b;
            if ((NOUT % 16 == 0) || (col < NOUT)) {
                b.x = W[(size_t)kb * NOUT + col];
                b.y = W[(size_t)(kb + 1) * NOUT + col];
            } else {
                b.x = 0.0f;
                b.y = 0.0f;
            }
            acc[t] = __builtin_amdgcn_wmma_f32_16x16x4_f32(
                false, a, false, b, (short)0, acc[t], false, false);
        }
    }

#pragma unroll
    for (int t = 0; t < NT; ++t) {
        const int col = t * 16 + l15;
        if ((NOUT % 16 != 0) && (col >= NOUT)) continue;
        const float bv = bias[col];
#pragma unroll
        for (int r = 0; r < 8; ++r) {
            const int m = wave * 16 + r + half * 8;
            H[(size_t)m * NOUT + col] = acc[t][r] + bv;
        }
    }
}

// ---------------------------------------------------------------------------
// Host-side launch
// ---------------------------------------------------------------------------
extern "C" void kernel_launch(void* const* d_in, const int* in_sizes, int n_in,
                              void* d_out, int out_size, void* d_ws, size_t ws_size,
                              hipStream_t stream) {
    const float* x    = (const float*)d_in[0];
    const int*   eidx = (const int*)d_in[1];
    const int*   src  = eidx;              // edge_index[0]
    const int*   dst  = eidx + N_EDGES;    // edge_index[1]
    const float* W1 = (const float*)d_in[2];
    const float* b1 = (const float*)d_in[3];
    const float* g1 = (const float*)d_in[4];
    const float* be1 = (const float*)d_in[5];
    const float* W2 = (const float*)d_in[6];
    const float* b2 = (const float*)d_in[7];
    const float* g2 = (const float*)d_in[8];
    const float* be2 = (const float*)d_in[9];
    const float* W3 = (const float*)d_in[10];
    const float* b3 = (const float*)d_in[11];
    float* out = (float*)d_out;

    // workspace layout (floats)
    float* ws      = (float*)d_ws;
    float* hbuf    = ws;                         // N*64
    float* aggbuf  = hbuf + N_NODES * HID_C;     // N*64
    float* en      = aggbuf + N_NODES * HID_C;   // E
    float* dis     = en + N_EDGES;               // N
    float* sn      = dis + N_NODES;              // N
    float* deg     = sn + N_NODES;               // N
    float* stats   = deg + N_NODES;              // 128 (sum, sumsq)
    float* ss      = stats + 2 * HID_C;          // 128 (scale, shift)

    const int BLK = 256;
    const int gN   = (N_NODES + BLK - 1) / BLK;
    const int gE   = (N_EDGES + BLK - 1) / BLK;
    const int gEl64 = 8192;   // grid-stride for N*64 / E*64 elementwise & scatter
    const int nRowBlocks = N_NODES / 16;                      // 6250 (exact)
    const int gGemm = (nRowBlocks + (BLK / 32) - 1) / (BLK / 32);

    // --- degree + norms -----------------------------------------------------
    k_zero<<<gN, BLK, 0, stream>>>(deg, N_NODES);
    k_degree<<<gE, BLK, 0, stream>>>(dst, deg);
    k_node_norm<<<gN, BLK, 0, stream>>>(deg, dis, sn);
    k_edge_norm<<<gE, BLK, 0, stream>>>(src, dst, dis, en);

    // --- layer 1: GCNConv(128->64) + BN + ReLU ------------------------------
    k_gemm_wmma<IN_C, HID_C><<<gGemm, BLK, 0, stream>>>(x, W1, b1, hbuf, nRowBlocks);
    k_init_scaled<HID_C><<<gEl64, BLK, 0, stream>>>(hbuf, sn, aggbuf);
    k_scatter_add<HID_C><<<gEl64, BLK, 0, stream>>>(hbuf, src, dst, en, aggbuf);
    k_zero<<<1, BLK, 0, stream>>>(stats, 2 * HID_C);
    k_bn_stats<<<1024, BLK, 0, stream>>>(aggbuf, stats);
    k_bn_finalize<<<1, 64, 0, stream>>>(stats, g1, be1, ss);
    k_bn_apply_relu<<<gEl64, BLK, 0, stream>>>(aggbuf, ss);

    // --- layer 2: GCNConv(64->64) + BN + ReLU -------------------------------
    k_gemm_wmma<HID_C, HID_C><<<gGemm, BLK, 0, stream>>>(aggbuf, W2, b2, hbuf, nRowBlocks);
    k_init_scaled<HID_C><<<gEl64, BLK, 0, stream>>>(hbuf, sn, aggbuf);   // aggbuf consumed by GEMM2 already
    k_scatter_add<HID_C><<<gEl64, BLK, 0, stream>>>(hbuf, src, dst, en, aggbuf);
    k_zero<<<1, BLK, 0, stream>>>(stats, 2 * HID_C);
    k_bn_stats<<<1024, BLK, 0, stream>>>(aggbuf, stats);
    k_bn_finalize<<<1, 64, 0, stream>>>(stats, g2, be2, ss);
    k_bn_apply_relu<<<gEl64, BLK, 0, stream>>>(aggbuf, ss);

    // --- layer 3: GCNConv(64->40) -> d_out ----------------------------------
    k_gemm_wmma<HID_C, OUT_C><<<gGemm, BLK, 0, stream>>>(aggbuf, W3, b3, hbuf, nRowBlocks);
    k_init_scaled<OUT_C><<<gEl64, BLK, 0, stream>>>(hbuf, sn, out);
    k_scatter_add<OUT_C><<<gEl64, BLK, 0, stream>>>(hbuf, src, dst, en, out);
}